// HistogramLayer_91087666413575
// MI455X (gfx1250) — compile-verified
//
#include <hip/hip_runtime.h>

// HistogramLayer: per-pixel argmax over 8 channels -> one-hot * sqrt(g0^2+g1^2).
// Memory-bound streaming kernel: 72 B/px, ~680 MB total -> ~29 us at 23.3 TB/s.
// Strategy: float4 (B128) coalesced non-temporal loads/stores, branch-free selects.

typedef float f4 __attribute__((ext_vector_type(4)));

__global__ __launch_bounds__(256) void hist_layer_kernel(
    const float* __restrict__ x, float* __restrict__ out, long long mn) {
  const long long v    = (long long)blockIdx.x * blockDim.x + threadIdx.x;
  const long long base = v * 4;
  if (base >= mn) return;

  const float* px = x + base;

  if (base + 4 <= mn) {
    // ---- vector fast path: 4 contiguous pixels, 10x global_load_b128 ----
    f4 c[8];
#pragma unroll
    for (int ch = 0; ch < 8; ++ch)
      c[ch] = __builtin_nontemporal_load(
          (const f4*)(px + (long long)ch * mn));
    const f4 g0 = __builtin_nontemporal_load((const f4*)(px + 8LL * mn));
    const f4 g1 = __builtin_nontemporal_load((const f4*)(px + 9LL * mn));

    int bi[4];
    f4  mag;
#pragma unroll
    for (int j = 0; j < 4; ++j) {
      float best = c[0][j];
      int   b    = 0;
#pragma unroll
      for (int ch = 1; ch < 8; ++ch) {
        const float val = c[ch][j];
        if (val > best) { best = val; b = ch; }  // strict > => first-max wins
      }
      bi[j]  = b;
      mag[j] = sqrtf(g0[j] * g0[j] + g1[j] * g1[j]);
    }

    float* po = out + base;
#pragma unroll
    for (int ch = 0; ch < 8; ++ch) {
      f4 o;
#pragma unroll
      for (int j = 0; j < 4; ++j) o[j] = (bi[j] == ch) ? mag[j] : 0.0f;
      __builtin_nontemporal_store(o, (f4*)(po + (long long)ch * mn));
    }
  } else {
    // ---- scalar tail (unused for 3072x3072, kept for generality) ----
    for (long long p = base; p < mn; ++p) {
      const float* q    = x + p;
      float        best = q[0];
      int          b    = 0;
      for (int ch = 1; ch < 8; ++ch) {
        const float val = q[(long long)ch * mn];
        if (val > best) { best = val; b = ch; }
      }
      const float a0 = q[8LL * mn], a1 = q[9LL * mn];
      const float m  = sqrtf(a0 * a0 + a1 * a1);
      for (int ch = 0; ch < 8; ++ch)
        out[p + (long long)ch * mn] = (ch == b) ? m : 0.0f;
    }
  }
}

extern "C" void kernel_launch(void* const* d_in, const int* in_sizes, int n_in,
                              void* d_out, int out_size, void* d_ws, size_t ws_size,
                              hipStream_t stream) {
  const float* x   = (const float*)d_in[0];
  float*       out = (float*)d_out;

  const long long total = (long long)in_sizes[0]; // 1*10*M*N
  const long long mn    = total / 10;             // M*N pixels
  const long long nvec  = (mn + 3) / 4;           // float4 groups

  const int       block = 256;                    // 8 wave32 waves
  const long long grid  = (nvec + block - 1) / block;

  hist_layer_kernel<<<dim3((unsigned int)grid), dim3(block), 0, stream>>>(x, out, mn);
}